// SSNN_75969381531906
// MI455X (gfx1250) — compile-verified
//
#include <hip/hip_runtime.h>

// ---------------------------------------------------------------------------
// SSM scan:  x_t = A x_{t-1} + B u_t ;  y_t = C x_t + D u_t
// u: (32, 4096, 512) f32   A: (1024,1024)  B: (1024,512)
// C: (512,1024)  D: (512,512)   out y: (32, 4096, 512) f32
//
// 2 batch-groups of 16 columns (WMMA N=16). 16 cooperating workgroups per
// group (64 x-rows / 32 y-rows each), persistent over all 4096 steps,
// synchronized once per step through per-step arrive counters in d_ws.
// All matmuls use V_WMMA_F32_16X16X4_F32 (fp32, matches reference precision).
// The step-invariant A row-block (256 KB) is staged once into LDS (320 KB on
// CDNA5) with a bank-conflict-free padded stride; B/C/D stay L2-resident.
// ---------------------------------------------------------------------------

#define XL 1024
#define UL 512
#define YL 512
#define SEQ 4096
#define NB 16           // batch columns per group == WMMA N
#define NGROUPS 2
#define WPG 16          // workgroups cooperating per group
#define THREADS 256     // 8 waves (wave32)

#define AROWS 64                  // x rows per WG
#define ASTRIDE 1028              // padded LDS row stride (banks spread by 4)
#define LDS_A_FLOATS (AROWS * ASTRIDE)
#define LDS_CBUF_FLOATS (8 * 32 * 8)
#define SMEM_BYTES ((LDS_A_FLOATS + LDS_CBUF_FLOATS) * 4)

typedef float v2f __attribute__((ext_vector_type(2)));
typedef float v4f __attribute__((ext_vector_type(4)));
typedef float v8f __attribute__((ext_vector_type(8)));

__device__ __forceinline__ v8f wmma4(v2f a, v2f b, v8f c) {
  // D = A(16x4) * B(4x16) + C(16x16), all fp32
  return __builtin_amdgcn_wmma_f32_16x16x4_f32(
      /*neg_a=*/false, a, /*neg_b=*/false, b,
      /*c_mod=*/(short)0, c, /*reuse_a=*/false, /*reuse_b=*/false);
}

// K-slab GEMM microkernel: acc += Arows[16 x (K1-K0)] * Bcols[(K1-K0) x 16]
// arow/bcol already include the per-lane row/column offset; kk pattern
// follows the 16x16x4 fragment layout (half = lane>>4 picks K pair).
template <int K0, int K1>
__device__ __forceinline__ v8f gemm_slab(const float* __restrict__ arow,
                                         const float* __restrict__ bcol,
                                         int half, v8f acc) {
  #pragma unroll 8
  for (int k = K0; k < K1; k += 4) {
    const int kk = k + 2 * half;
    v2f a = *(const v2f*)(arow + kk);
    v2f b = *(const v2f*)(bcol + kk);
    acc = wmma4(a, b, acc);
  }
  return acc;
}

__global__ void init_ws(int* __restrict__ p, int n) {
  int i = blockIdx.x * blockDim.x + threadIdx.x;
  if (i < n) p[i] = 0;
}

__global__ __launch_bounds__(THREADS, 1)
void ssm_kernel(const float* __restrict__ u,  const float* __restrict__ A,
                const float* __restrict__ Bm, const float* __restrict__ Cm,
                const float* __restrict__ Dm, float* __restrict__ y,
                float* __restrict__ xstate, int* __restrict__ bars) {
  extern __shared__ float smem[];
  float* ldsA = smem;                       // [AROWS][ASTRIDE]
  float* cbuf = smem + LDS_A_FLOATS;        // [8 waves][32 lanes][8]

  const int wg    = blockIdx.x % WPG;
  const int group = blockIdx.x / WPG;
  const int bbase = group * NB;
  const int tid   = threadIdx.x;
  const int lane  = tid & 31;
  const int half  = lane >> 4;   // K-pair select in fragments
  const int l16   = lane & 15;

  // wave id as a *scalar*: role branches -> s_cbranch, literal trip counts,
  // EXEC stays all-ones around every WMMA.
  const int wave  = __builtin_amdgcn_readfirstlane(tid >> 5);

  const int xrow0 = wg * (XL / WPG);   // 64 rows of x per WG
  const int yrow0 = wg * (YL / WPG);   // 32 rows of y per WG

  // state buffers: [group][2][NB][XL]  (column-major so B-frags are float2)
  float* xbuf0 = xstate + (size_t)group * 2 * NB * XL;
  int*   bar   = bars + group * SEQ;

  // phase-X role: 4 tiles of 16 rows, wave pair splits K at 768
  const int xtile = wave & 3;
  const int kpart = wave >> 2;
  // phase-Y role: 2 tiles of 16 rows, 4-way K split (C:384/384/256+D:128/384)
  const int q     = wave & 3;
  const int yrow  = yrow0 + (wave >> 2) * 16;

  // ---- one-time: stage this WG's A row-block into LDS (coalesced) ----
  #pragma unroll 4
  for (int i = tid; i < AROWS * XL; i += THREADS) {
    const int r = i >> 10;            // local row 0..63
    const int c = i & (XL - 1);
    ldsA[r * ASTRIDE + c] = A[(size_t)(xrow0 + r) * XL + c];
  }
  __syncthreads();

  const float* uB    = u + (size_t)(bbase + l16) * SEQ * UL;
  const float* ArowL = ldsA + (size_t)(xtile * 16 + l16) * ASTRIDE;
  const float* Brow  = Bm + (size_t)(xrow0 + xtile * 16 + l16) * UL;
  const float* Crow  = Cm + (size_t)(yrow + l16) * XL;
  const float* Drow  = Dm + (size_t)(yrow + l16) * UL;

  for (int t = 0; t < SEQ; ++t) {
    const float* prev = xbuf0 + (size_t)(t & 1) * NB * XL;
    float*       cur  = xbuf0 + (size_t)((t + 1) & 1) * NB * XL;
    const float* ut   = uB + (size_t)t * UL;

    // hint-prefetch next step's u row for this lane's batch column
    if (t + 1 < SEQ)
      __builtin_prefetch(ut + UL + half * 256, 0, 0);

    // ---------------- Phase X: cur rows = A@prev + B@u_t -----------------
    v8f acc = {0.f,0.f,0.f,0.f,0.f,0.f,0.f,0.f};
    {
      const float* Xcol = prev + (size_t)l16 * XL;
      if (kpart == 0) {
        acc = gemm_slab<0, 768>(ArowL, Xcol, half, acc);     // A frags: LDS
      } else {
        acc = gemm_slab<768, 1024>(ArowL, Xcol, half, acc);  // A frags: LDS
        acc = gemm_slab<0, UL>(Brow, ut, half, acc);
        #pragma unroll
        for (int r = 0; r < 8; ++r) cbuf[(wave * 32 + lane) * 8 + r] = acc[r];
      }
    }
    __syncthreads();
    if (kpart == 0) {
      #pragma unroll
      for (int r = 0; r < 8; ++r) acc[r] += cbuf[((wave + 4) * 32 + lane) * 8 + r];
      // D-frag: vgpr r holds (row = xrow + r + 8*half, col = l16); 8 floats
      // per lane are contiguous -> two b128 stores.
      float* outp = cur + (size_t)l16 * XL + xrow0 + xtile * 16 + 8 * half;
      *(v4f*)(outp)     = v4f{acc[0], acc[1], acc[2], acc[3]};
      *(v4f*)(outp + 4) = v4f{acc[4], acc[5], acc[6], acc[7]};
    }

    // -------- per-step group barrier across the 16 cooperating WGs -------
    __threadfence();          // flush cur stores to device scope
    __syncthreads();
    if (tid == 0) {
      __hip_atomic_fetch_add(&bar[t], 1, __ATOMIC_ACQ_REL,
                             __HIP_MEMORY_SCOPE_AGENT);
      while (__hip_atomic_load(&bar[t], __ATOMIC_ACQUIRE,
                               __HIP_MEMORY_SCOPE_AGENT) < WPG) {
        __builtin_amdgcn_s_sleep(1);
      }
    }
    __syncthreads();
    __threadfence();          // invalidate near caches: see remote cur rows

    // ---------------- Phase Y: y_t = C@cur + D@u_t -----------------------
    v8f yacc = {0.f,0.f,0.f,0.f,0.f,0.f,0.f,0.f};
    {
      const float* Xcol = cur + (size_t)l16 * XL;
      if (q == 0) {
        yacc = gemm_slab<0, 384>(Crow, Xcol, half, yacc);
      } else if (q == 1) {
        yacc = gemm_slab<384, 768>(Crow, Xcol, half, yacc);
      } else if (q == 2) {
        yacc = gemm_slab<768, 1024>(Crow, Xcol, half, yacc);
        yacc = gemm_slab<0, 128>(Drow, ut, half, yacc);
      } else {
        yacc = gemm_slab<128, 512>(Drow, ut, half, yacc);
      }
    }
    if (q != 0) {
      #pragma unroll
      for (int r = 0; r < 8; ++r) cbuf[(wave * 32 + lane) * 8 + r] = yacc[r];
    }
    __syncthreads();
    if (q == 0) {
      #pragma unroll
      for (int r = 0; r < 8; ++r)
        yacc[r] += cbuf[((wave + 1) * 32 + lane) * 8 + r] +
                   cbuf[((wave + 2) * 32 + lane) * 8 + r] +
                   cbuf[((wave + 3) * 32 + lane) * 8 + r];
      float* yp = y + ((size_t)(bbase + l16) * SEQ + t) * YL + yrow + 8 * half;
      *(v4f*)(yp)     = v4f{yacc[0], yacc[1], yacc[2], yacc[3]};
      *(v4f*)(yp + 4) = v4f{yacc[4], yacc[5], yacc[6], yacc[7]};
    }
    __syncthreads();   // cbuf reuse next iteration
  }
}

extern "C" void kernel_launch(void* const* d_in, const int* in_sizes, int n_in,
                              void* d_out, int out_size, void* d_ws,
                              size_t ws_size, hipStream_t stream) {
  const float* u  = (const float*)d_in[0];
  const float* A  = (const float*)d_in[1];
  const float* Bm = (const float*)d_in[2];
  const float* Cm = (const float*)d_in[3];
  const float* Dm = (const float*)d_in[4];
  float* y = (float*)d_out;

  float* xstate = (float*)d_ws;                       // 256 KB state buffers
  int*   bars   = (int*)((char*)d_ws +
                         (size_t)NGROUPS * 2 * NB * XL * sizeof(float));

  const int ninit = NGROUPS * 2 * NB * XL + NGROUPS * SEQ;   // zero both
  init_ws<<<(ninit + 255) / 256, 256, 0, stream>>>((int*)d_ws, ninit);

  ssm_kernel<<<dim3(NGROUPS * WPG), dim3(THREADS), SMEM_BYTES, stream>>>(
      u, A, Bm, Cm, Dm, y, xstate, bars);
}